// Dynamics_33294586479064
// MI455X (gfx1250) — compile-verified
//
#include <hip/hip_runtime.h>
#include <math.h>

// Batched Kalman filter step for MI455X (gfx1250, wave32).
// One wave32 per batch element. All 16x16 dense products via
// V_WMMA_F32_16X16X4_F32 (fp32 WMMA, K=4), staged in LDS. The 32x32 SPD
// solve (pinv(S) for SPD S == inv(S)) is done as Cholesky + 2 triangular
// solves with lane-parallel LDS code.

#define XD 16
#define UD 8
#define AD 32
#define WPB 4          // waves per block
#define LDP 17         // padded leading dim for 16-col LDS tiles
#define LDSS 33        // padded leading dim for 32x32 S

typedef float v2f __attribute__((ext_vector_type(2)));
typedef float v8f __attribute__((ext_vector_type(8)));

struct WaveWS {
  float P1 [XD * LDP];   // symmetrized predicted covariance
  float TMP[XD * LDP];   // scratch 16x16 (A*cov, then KC/G)
  float T  [AD * LDP];   // scratch: raw A cov A^T (rows 0..15), then C*P1 (32x16), then (I-KC)P1
  float S  [AD * LDSS];  // innovation covariance / Cholesky factor L
  float X  [AD * LDP];   // K^T = S^{-1} (C P1)   (32x16)
  float VEC[64];         // [0:16) m1, [16:48) innov
};

__device__ __forceinline__ float softplusf(float x) {
  return (x > 20.0f) ? x : log1pf(__expf(x));
}

// ---- WMMA 16x16x4 fp32 operand loads (wave32 layouts per CDNA5 ISA 7.12.2) ----
// A (16x4): lane holds row M=lane%16, K = g + 2*(lane/16)
// B (4x16): lane holds col N=lane%16, K = g + 2*(lane/16)
// C/D     : lane holds col N=lane%16, rows M = v + 8*(lane/16)
__device__ __forceinline__ v2f ldA(const float* s, int ld, int k0, int lm, int half) {
  const float* p = s + lm * ld + k0 + 2 * half;
  v2f r; r.x = p[0]; r.y = p[1]; return r;
}
__device__ __forceinline__ v2f ldAT(const float* s, int ld, int k0, int lm, int half) {
  int k = k0 + 2 * half;
  v2f r; r.x = s[k * ld + lm]; r.y = s[(k + 1) * ld + lm]; return r;
}
__device__ __forceinline__ v2f ldB(const float* s, int ld, int k0, int lm, int half) {
  int k = k0 + 2 * half;
  v2f r; r.x = s[k * ld + lm]; r.y = s[(k + 1) * ld + lm]; return r;
}
__device__ __forceinline__ v2f ldBT(const float* s, int ld, int k0, int lm, int half) {
  const float* p = s + lm * ld + k0 + 2 * half;
  v2f r; r.x = p[0]; r.y = p[1]; return r;
}
__device__ __forceinline__ void stCD(float* d, int ld, v8f c, int lm, int half) {
#pragma unroll
  for (int v = 0; v < 8; ++v) d[(v + 8 * half) * ld + lm] = c[v];
}

// D = op(A) * op(B), 16xKD * KDx16, accumulated over K in steps of 4.
template <bool TA, bool TB, int KD>
__device__ __forceinline__ v8f mm16(const float* A, int lda, const float* B, int ldb,
                                    int lm, int half) {
  v8f acc = {0.f, 0.f, 0.f, 0.f, 0.f, 0.f, 0.f, 0.f};
#pragma unroll
  for (int k0 = 0; k0 < KD; k0 += 4) {
    v2f a, b;
    if constexpr (TA) a = ldAT(A, lda, k0, lm, half); else a = ldA(A, lda, k0, lm, half);
    if constexpr (TB) b = ldBT(B, ldb, k0, lm, half); else b = ldB(B, ldb, k0, lm, half);
    acc = __builtin_amdgcn_wmma_f32_16x16x4_f32(false, a, false, b, (short)0, acc,
                                                false, false);
  }
  return acc;
}

__global__ __launch_bounds__(WPB * 32) void kf_fused(
    const float* __restrict__ mean, const float* __restrict__ cov,
    const float* __restrict__ uin,  const float* __restrict__ aobs,
    const float* __restrict__ Ag,   const float* __restrict__ Bg,
    const float* __restrict__ Cg,   const float* __restrict__ nxg,
    const float* __restrict__ nag,  float* __restrict__ m2out,
    float* __restrict__ p2out) {
  __shared__ WaveWS ws[WPB];
  const int tid  = threadIdx.x;
  const int wv   = tid >> 5;
  const int lane = tid & 31;
  const int half = lane >> 4;
  const int lm   = lane & 15;
  const int b    = blockIdx.x * WPB + wv;

  WaveWS& W = ws[wv];
  const float* covb = cov + (size_t)b * (XD * XD);

  // ---- m1 = mean A^T + u B^T (lanes 0..15, one state row each) ----
  if (lane < XD) {
    float s = 0.f;
#pragma unroll
    for (int j = 0; j < XD; ++j) s += Ag[lane * XD + j] * mean[(size_t)b * XD + j];
#pragma unroll
    for (int j = 0; j < UD; ++j) s += Bg[lane * UD + j] * uin[(size_t)b * UD + j];
    W.VEC[lane] = s;
  }
  __syncthreads();

  // ---- innov = a_obs - C m1 (all 32 lanes, one obs row each) ----
  {
    float s = aobs[(size_t)b * AD + lane];
#pragma unroll
    for (int j = 0; j < XD; ++j) s -= Cg[lane * XD + j] * W.VEC[j];
    W.VEC[16 + lane] = s;
  }

  // ---- TMP = A * cov ----
  stCD(W.TMP, LDP, mm16<false, false, XD>(Ag, XD, covb, XD, lm, half), lm, half);
  __syncthreads();
  // ---- T[0:16] = (A*cov) * A^T  (raw, unsymmetrized) ----
  stCD(W.T, LDP, mm16<false, true, XD>(W.TMP, LDP, Ag, XD, lm, half), lm, half);
  __syncthreads();
  // ---- P1 = 0.5(M+M^T) + diag(softplus(nx)+1e-4+1e-6) ----
  {
    float dsp = softplusf(nxg[lm]) + 1e-4f + 1e-6f;
#pragma unroll
    for (int v = 0; v < 8; ++v) {
      int r = v + 8 * half;
      float val = 0.5f * (W.T[r * LDP + lm] + W.T[lm * LDP + r]);
      if (r == lm) val += dsp;
      W.P1[r * LDP + lm] = val;
    }
  }
  __syncthreads();

  // ---- T = C * P1 (32x16). Since P1 symmetric, T^T = P1 C^T = PCt. ----
#pragma unroll
  for (int i = 0; i < 2; ++i)
    stCD(W.T + i * 16 * LDP, LDP,
         mm16<false, false, XD>(Cg + i * 16 * XD, XD, W.P1, LDP, lm, half), lm, half);
  __syncthreads();

  // ---- S = T * C^T + diag(softplus(na)+1e-4)  (four 16x16 tiles) ----
#pragma unroll
  for (int i = 0; i < 2; ++i)
#pragma unroll
    for (int j = 0; j < 2; ++j)
      stCD(W.S + i * 16 * LDSS + j * 16, LDSS,
           mm16<false, true, XD>(W.T + i * 16 * LDP, LDP, Cg + j * 16 * XD, XD, lm, half),
           lm, half);
  __syncthreads();
  W.S[lane * LDSS + lane] += softplusf(nag[lane]) + 1e-4f;

  // ---- Cholesky: S (lower triangle) -> L, lane-parallel over rows ----
  for (int k = 0; k < AD; ++k) {
    __syncthreads();
    float d   = W.S[k * LDSS + k];       // broadcast read (issued before lane-k store)
    float lkk = sqrtf(d);
    float inv = 1.0f / lkk;
    if (lane == k) W.S[k * LDSS + k] = lkk;
    if (lane > k)  W.S[lane * LDSS + k] *= inv;
    __syncthreads();
    if (lane > k) {
      float lik = W.S[lane * LDSS + k];
      for (int j2 = k + 1; j2 <= lane; ++j2)
        W.S[lane * LDSS + j2] -= lik * W.S[j2 * LDSS + k];
    }
  }
  __syncthreads();

  // ---- Solve S X = T  (X = K^T, 32x16). Lane c owns RHS column c: no barriers. ----
  if (lane < XD) {
    const int c = lane;
    for (int i = 0; i < AD; ++i) {           // forward: L Y = T
      float s = W.T[i * LDP + c];
      for (int m = 0; m < i; ++m) s -= W.S[i * LDSS + m] * W.X[m * LDP + c];
      W.X[i * LDP + c] = s / W.S[i * LDSS + i];
    }
    for (int i = AD - 1; i >= 0; --i) {      // backward: L^T X = Y (in place)
      float s = W.X[i * LDP + c];
      for (int m = i + 1; m < AD; ++m) s -= W.S[m * LDSS + i] * W.X[m * LDP + c];
      W.X[i * LDP + c] = s / W.S[i * LDSS + i];
    }
  }
  __syncthreads();

  // ---- TMP = K*C = X^T * C  (K=32 accumulation) ----
  stCD(W.TMP, LDP, mm16<true, false, AD>(W.X, LDP, Cg, XD, lm, half), lm, half);
  // ---- G = I - KC (each lane rewrites exactly the elements it just stored) ----
#pragma unroll
  for (int v = 0; v < 8; ++v) {
    int r = v + 8 * half;
    W.TMP[r * LDP + lm] = ((r == lm) ? 1.0f : 0.0f) - W.TMP[r * LDP + lm];
  }
  __syncthreads();
  // ---- T[0:16] = G * P1 (raw P2) ----
  stCD(W.T, LDP, mm16<false, false, XD>(W.TMP, LDP, W.P1, LDP, lm, half), lm, half);
  __syncthreads();

  // ---- P2 = 0.5(M+M^T) + 1e-6 I  -> global ----
  {
    float* p2b = p2out + (size_t)b * (XD * XD);
#pragma unroll
    for (int v = 0; v < 8; ++v) {
      int r = v + 8 * half;
      float val = 0.5f * (W.T[r * LDP + lm] + W.T[lm * LDP + r]);
      if (r == lm) val += 1e-6f;
      p2b[r * XD + lm] = val;
    }
  }
  // ---- m2 = m1 + K innov  (K[x][a] = X[a][x]) ----
  if (lane < XD) {
    float s = W.VEC[lane];
#pragma unroll
    for (int a = 0; a < AD; ++a) s += W.X[a * LDP + lane] * W.VEC[16 + a];
    m2out[(size_t)b * XD + lane] = s;
  }
}

extern "C" void kernel_launch(void* const* d_in, const int* in_sizes, int n_in,
                              void* d_out, int out_size, void* d_ws, size_t ws_size,
                              hipStream_t stream) {
  const float* mean = (const float*)d_in[0];
  const float* cov  = (const float*)d_in[1];
  const float* u    = (const float*)d_in[2];
  const float* aobs = (const float*)d_in[3];
  const float* A    = (const float*)d_in[4];
  const float* B    = (const float*)d_in[5];
  const float* C    = (const float*)d_in[6];
  const float* nx   = (const float*)d_in[7];
  const float* na   = (const float*)d_in[8];

  const int batch = in_sizes[0] / XD;  // 32768; divisible by WPB
  float* m2 = (float*)d_out;
  float* P2 = m2 + (size_t)batch * XD;

  dim3 grid(batch / WPB), block(WPB * 32);
  hipLaunchKernelGGL(kf_fused, grid, block, 0, stream,
                     mean, cov, u, aobs, A, B, C, nx, na, m2, P2);
}